// SpiralConv_89472758710752
// MI455X (gfx1250) — compile-verified
//
#include <hip/hip_runtime.h>
#include <hip/hip_bf16.h>

// SpiralConv: out[b,n,o] = sum_f feat[b,n,f] * W[o,f] + bias[o]
//   feat[b,n, s*32+c] = x[b, idx[n,s], c]   (s = 0..8)
//   feat[b,n, 288+c]  = max_n x[b,n,c]
// BS=16, N_NODES=50000, SEQ=9, IN_C=OUT_C=32, F=320.

typedef float v2f __attribute__((ext_vector_type(2)));
typedef float v8f __attribute__((ext_vector_type(8)));

#define BS_      16
#define SEQ_     9
#define INC_     32
#define OUTC_    32
#define F_       320      // (SEQ+1)*INC
#define TN_      128      // nodes per block (8 wave32 M-tiles of 16)
#define THREADS_ 256
#define FPAD_    (F_ + 4) // LDS row stride pad -> conflict-free ds_load_b64

// ---- monotonic float <-> uint mapping for atomic max reduction ----
__device__ __forceinline__ unsigned enc_f32(float f) {
  unsigned u = __float_as_uint(f);
  return (u & 0x80000000u) ? ~u : (u | 0x80000000u);
}
__device__ __forceinline__ float dec_f32(unsigned m) {
  unsigned u = (m & 0x80000000u) ? (m ^ 0x80000000u) : ~m;
  return __uint_as_float(u);
}

// Pass 0: init encoded gmax workspace to enc(-inf) (ws is poisoned; must be
// re-initialized deterministically every launch).
__global__ void gmax_init_kernel(unsigned* __restrict__ g) {
  int t = blockIdx.x * blockDim.x + threadIdx.x;
  if (t < BS_ * INC_) g[t] = enc_f32(-INFINITY);
}

// Pass 1: gmax[b][c] = max over nodes of x[b,n,c], via encoded atomicMax.
__global__ void gmax_kernel(const float* __restrict__ x,
                            unsigned* __restrict__ g,
                            int n_nodes, int chunk) {
  int b   = blockIdx.y;
  int c   = threadIdx.x & 31;
  int grp = threadIdx.x >> 5;                 // 8 node groups per block
  int start = blockIdx.x * chunk;
  int end   = start + chunk;
  if (end > n_nodes) end = n_nodes;
  const float* xb = x + (size_t)b * n_nodes * INC_;
  float m = -INFINITY;
  for (int n = start + grp; n < end; n += 8)
    m = fmaxf(m, xb[(size_t)n * INC_ + c]);
  atomicMax(&g[b * INC_ + c], enc_f32(m));
}

// Pass 2: WMMA GEMM. One block = one batch x 128-node tile; wave w owns the
// 16-node M-tile [w*16, w*16+16) and both 16x16 N-tiles (OUT_C = 32).
__global__ __launch_bounds__(THREADS_) void spiral_wmma_kernel(
    const float* __restrict__ x, const int* __restrict__ idx,
    const float* __restrict__ W, const float* __restrict__ bias,
    const unsigned* __restrict__ genc, float* __restrict__ out, int n_nodes) {
  __shared__ __align__(16) float sW[OUTC_][FPAD_]; // 32 x 324 f32 = 41.5 KB
  __shared__ int   sRow[TN_][SEQ_];                // gathered row ids (4.6 KB)
  __shared__ __align__(8) float sG[INC_];          // decoded gmax row

  const int t        = threadIdx.x;
  const int b        = blockIdx.y;
  const int nodeBase = blockIdx.x * TN_;

  // Stage W (row-major [32][320]) into padded LDS, float4 chunks.
  {
    const float4* W4 = (const float4*)W;
    for (int i = t; i < OUTC_ * (F_ / 4); i += THREADS_) {
      int o = i / (F_ / 4);
      int q = i % (F_ / 4);
      *(float4*)&sW[o][q * 4] = W4[i];
    }
  }
  if (t < INC_) sG[t] = dec_f32(genc[b * INC_ + t]);
  for (int j = t; j < TN_ * SEQ_; j += THREADS_) {
    int m = j / SEQ_, s = j % SEQ_;
    int node = nodeBase + m;
    if (node >= n_nodes) node = n_nodes - 1;   // clamp; tail stores are guarded
    sRow[m][s] = idx[node * SEQ_ + s];
  }
  __syncthreads();

  const int wave = t >> 5;        // 0..7 -> M-tile
  const int lane = t & 31;
  const int half = lane >> 4;     // A: K-pair selector (K0+0,1 vs K0+2,3)
  const int lm   = lane & 15;     // A: row M; B/D: column N
  const int m0   = wave * 16;

  const float* xb = x + (size_t)b * n_nodes * INC_;

  v8f acc0 = {};                  // D tile, N = 0..15
  v8f acc1 = {};                  // D tile, N = 16..31

  // K = 0..287: gathered part of feat. s = K0/32 is lane-uniform (scalar
  // branch-free), A fragment = 8B global load, L2-resident gather.
  #pragma unroll 4
  for (int kk = 0; kk < (SEQ_ * INC_) / 4; ++kk) {
    const int K0 = kk * 4;
    const int s  = K0 >> 5;
    const int c  = (K0 & 31) + 2 * half;
    const int row = sRow[m0 + lm][s];
    v2f a  = *(const v2f*)(xb + (size_t)row * INC_ + c);       // A[M=lm][K0+2h..+1]
    v2f b0 = *(const v2f*)&sW[lm][K0 + 2 * half];              // B[k][n]=W[n][K0+k]
    v2f b1 = *(const v2f*)&sW[16 + lm][K0 + 2 * half];
    acc0 = __builtin_amdgcn_wmma_f32_16x16x4_f32(false, a, false, b0,
                                                 (short)0, acc0, false, false);
    acc1 = __builtin_amdgcn_wmma_f32_16x16x4_f32(false, a, false, b1,
                                                 (short)0, acc1, false, false);
  }
  // K = 288..319: broadcast gmax row from LDS.
  #pragma unroll
  for (int kk = (SEQ_ * INC_) / 4; kk < F_ / 4; ++kk) {
    const int K0 = kk * 4;
    const int c  = (K0 & 31) + 2 * half;
    v2f a  = *(const v2f*)&sG[c];
    v2f b0 = *(const v2f*)&sW[lm][K0 + 2 * half];
    v2f b1 = *(const v2f*)&sW[16 + lm][K0 + 2 * half];
    acc0 = __builtin_amdgcn_wmma_f32_16x16x4_f32(false, a, false, b0,
                                                 (short)0, acc0, false, false);
    acc1 = __builtin_amdgcn_wmma_f32_16x16x4_f32(false, a, false, b1,
                                                 (short)0, acc1, false, false);
  }

  // D layout: VGPR r, lane l -> M = r + 8*(l>=16), N = l&15.
  const float bo0 = bias[lm];
  const float bo1 = bias[16 + lm];
  #pragma unroll
  for (int r = 0; r < 8; ++r) {
    int node = nodeBase + m0 + r + 8 * half;
    if (node < n_nodes) {
      float* orow = out + ((size_t)b * n_nodes + node) * OUTC_;
      orow[lm]      = acc0[r] + bo0;
      orow[16 + lm] = acc1[r] + bo1;
    }
  }
}

extern "C" void kernel_launch(void* const* d_in, const int* in_sizes, int n_in,
                              void* d_out, int out_size, void* d_ws, size_t ws_size,
                              hipStream_t stream) {
  const float* x    = (const float*)d_in[0];
  const int*   idx  = (const int*)d_in[1];   // JAX default x64-off -> int32
  const float* W    = (const float*)d_in[2];
  const float* bias = (const float*)d_in[3];
  float*       out  = (float*)d_out;
  unsigned*    genc = (unsigned*)d_ws;       // 16*32 encoded maxes (2 KB)

  const int n_nodes = in_sizes[1] / SEQ_;    // indices is [n_nodes, 9]

  hipLaunchKernelGGL(gmax_init_kernel, dim3(1), dim3(512), 0, stream, genc);

  const int chunk = 2048;
  dim3 g1((n_nodes + chunk - 1) / chunk, BS_);
  hipLaunchKernelGGL(gmax_kernel, g1, dim3(256), 0, stream, x, genc, n_nodes, chunk);

  dim3 g2((n_nodes + TN_ - 1) / TN_, BS_);
  hipLaunchKernelGGL(spiral_wmma_kernel, g2, dim3(THREADS_), 0, stream,
                     x, idx, W, bias, genc, out, n_nodes);
}